// ParallelTransport_83270825935581
// MI455X (gfx1250) — compile-verified
//
#include <hip/hip_runtime.h>
#include <hip/hip_bf16.h>
#include <math.h>

// ---------------------------------------------------------------------------
// ParallelTransport, factored:
//   out[h,j,d] = (1/S) * sum_c Hsum[h,j,c] * W2K[h,j,c,d] + b2K[h,j,d]
//     Hsum[h,j,c] = sum_i tanh(QW1[h,i,c] + KW1[h,j,c] + b1[c])
//     W2K[h,j,c,d] = sum_e W2[c, d*64+e] * K[h,j,e]
//     b2K[h,j,d]   = sum_e b2[d*64+e]    * K[h,j,e]
// Sizes: H=2, S=128, D=64, 2D=128.  hj = h*128+j in [0,256).
// Work drops from ~34.4 GFLOP (naive) to ~0.28 GFLOP; W2 (2MB) is read 16x
// from L2 (192MB) and once from HBM -> latency-bound. With ~1 wave/SIMD the
// only latency hiding is intra-wave MLP, so kernel 3 double-buffers the
// A-fragments across c-iterations and requests a 1-wave/EU register budget.
// ---------------------------------------------------------------------------

typedef __attribute__((ext_vector_type(2))) float v2f;
typedef __attribute__((ext_vector_type(8))) float v8f;

#define HS 256      // H*S
#define C2D 128     // 2*D
#define DD 64       // D
#define NC_CHUNKS 8 // c-chunks of 16

// ---- Kernel 1: QW1[h,i,c] = Q[h,i,:]@W1[:64,c]; KW1[h,j,c] = K[h,j,:]@W1[64:,c]
__global__ void k_qkw1(const float* __restrict__ Q, const float* __restrict__ K,
                       const float* __restrict__ W1,
                       float* __restrict__ QW1, float* __restrict__ KW1) {
  int tid = blockIdx.x * blockDim.x + threadIdx.x;   // 32768 = HS*C2D
  int hi = tid >> 7;
  int c  = tid & 127;
  const float* q = Q + hi * 64;
  const float* k = K + hi * 64;
  float sq = 0.f, sk = 0.f;
  #pragma unroll 8
  for (int a = 0; a < 64; ++a) {
    sq += q[a] * W1[a * C2D + c];
    sk += k[a] * W1[(64 + a) * C2D + c];
  }
  QW1[tid] = sq;
  KW1[tid] = sk;
}

// ---- Kernel 2: Hsum[h,j,c] = sum_i tanh(QW1[h,i,c] + KW1[h,j,c] + b1[c])
__global__ void k_hsum(const float* __restrict__ QW1, const float* __restrict__ KW1,
                       const float* __restrict__ b1, float* __restrict__ Hsum) {
  int tid = blockIdx.x * blockDim.x + threadIdx.x;   // 32768 = HS*C2D
  int hj = tid >> 7;
  int c  = tid & 127;
  int h  = hj >> 7;
  float base = KW1[tid] + b1[c];
  const float* qrow = QW1 + (h * 128) * C2D + c;     // coalesced in c for fixed i
  float s = 0.f;
  for (int i = 0; i < 128; ++i)
    s += tanhf(qrow[i * C2D] + base);
  Hsum[tid] = s;
}

// ---- Kernel 3 (WMMA): partial[cc][hj,d] = sum_{c in chunk} Hsum[hj,c]*W2K[hj,c,d]
// GEMM core: D(16x16) += A(16x4) x B(4x16), f32, chained over e (K=64 -> 16 steps).
//   A rows   : m -> d = dt*16+m, A[m][k] = W2[c*4096 + d*64 + e]
//   B cols   : n -> hj = ct*16+n, B[k][n] = K[hj*64 + e]
// Frag layout (ISA 7.12.2): lane = half*16 + l16;
//   A: M=l16, K = e_base + 2*half + vgpr
//   B: N=l16, K = e_base + 2*half + vgpr   (mirrored)
//   D: N=l16, M = vgpr + 8*half
// Double-buffered: while iteration c's serially-dependent 16-wmma chain runs,
// iteration c+1's 16 A-frag loads are already in flight. amdgpu_waves_per_eu(1)
// relaxes the scheduler's register-pressure target so it stops collapsing the
// load buffer into a single load->wait(0)->wmma lockstep.
__global__ __launch_bounds__(32) __attribute__((amdgpu_waves_per_eu(1, 1)))
void k_w2k(const float* __restrict__ K, const float* __restrict__ W2,
           const float* __restrict__ Hsum, float* __restrict__ partial) {
  const int lane = threadIdx.x;
  const int ct = blockIdx.x;         // 0..15 : column tile over hj
  const int dt = blockIdx.y;         // 0..3  : row tile over d
  const int cc = blockIdx.z;         // 0..7  : c-chunk
  const int half = lane >> 4;
  const int l16  = lane & 15;
  const int k0   = half * 2;
  const int hj   = ct * 16 + l16;    // this lane's B/D column
  const int drow = dt * 16 + l16;    // this lane's A row (d)
  const int cbase = cc * 16;

  // Preload B fragments: invariant over c. bfrag[es].{x,y} = K[hj, es*4 + k0 + {0,1}]
  v2f bfrag[16];
  #pragma unroll
  for (int es = 0; es < 16; ++es)
    bfrag[es] = *(const v2f*)(K + hj * 64 + es * 4 + k0);

  // Prologue: load A-frags + Hsum scalar for first c.
  v2f cur[16];
  {
    const float* Arow = W2 + (size_t)cbase * 4096 + (size_t)drow * 64;
    #pragma unroll
    for (int es = 0; es < 16; ++es)
      cur[es] = *(const v2f*)(Arow + es * 4 + k0);
  }
  float hs = Hsum[hj * C2D + cbase];

  v8f outacc = {};
  #pragma unroll
  for (int cidx = 0; cidx < 16; ++cidx) {
    // Prefetch next iteration's A-frags while this chain computes.
    v2f nxt[16];
    float hs_n = 0.f;
    if (cidx < 15) {
      const float* ArowN = W2 + (size_t)(cbase + cidx + 1) * 4096 + (size_t)drow * 64;
      #pragma unroll
      for (int es = 0; es < 16; ++es)
        nxt[es] = *(const v2f*)(ArowN + es * 4 + k0);
      hs_n = Hsum[hj * C2D + cbase + cidx + 1];
    }

    // Serially-dependent wmma chain for current c.
    v8f acc = {};
    #pragma unroll
    for (int es = 0; es < 16; ++es)
      acc = __builtin_amdgcn_wmma_f32_16x16x4_f32(
          /*neg_a=*/false, cur[es], /*neg_b=*/false, bfrag[es],
          /*c_mod=*/(short)0, acc, /*reuse_a=*/false, /*reuse_b=*/false);

    // Scale the 16x16 W2K tile by Hsum[hj, c] (all of a lane's D values share
    // column hj, so this is a per-lane scalar multiply).
    #pragma unroll
    for (int v = 0; v < 8; ++v) outacc[v] += hs * acc[v];

    // Rotate buffers (vanishes under full unroll via SSA renaming).
    #pragma unroll
    for (int es = 0; es < 16; ++es) cur[es] = nxt[es];
    hs = hs_n;
  }

  // D-frag row m = v + 8*half -> d = dt*16 + m
  float* p = partial + (size_t)cc * (HS * DD) + (size_t)hj * DD + dt * 16 + 8 * half;
  #pragma unroll
  for (int v = 0; v < 8; ++v) p[v] = outacc[v];
}

// ---- Kernel 4: out[hj,d] = (1/S) * sum_cc partial + b2K[hj,d]
__global__ void k_out(const float* __restrict__ partial, const float* __restrict__ K,
                      const float* __restrict__ b2, float* __restrict__ out) {
  int tid = blockIdx.x * blockDim.x + threadIdx.x;   // 16384 = HS*DD
  int hj = tid >> 6;
  int d  = tid & 63;
  float s = 0.f;
  #pragma unroll
  for (int cc = 0; cc < NC_CHUNKS; ++cc)
    s += partial[cc * (HS * DD) + tid];
  const float* b2row = b2 + d * 64;
  const float* krow  = K + hj * 64;
  float b2k = 0.f;
  #pragma unroll 8
  for (int e = 0; e < 64; ++e)
    b2k += b2row[e] * krow[e];
  out[tid] = s * (1.f / 128.f) + b2k;
}

extern "C" void kernel_launch(void* const* d_in, const int* in_sizes, int n_in,
                              void* d_out, int out_size, void* d_ws, size_t ws_size,
                              hipStream_t stream) {
  const float* Q  = (const float*)d_in[0];   // [1,2,128,64]
  const float* K  = (const float*)d_in[1];   // [1,2,128,64]
  const float* W1 = (const float*)d_in[2];   // [128,128]
  const float* b1 = (const float*)d_in[3];   // [128]
  const float* W2 = (const float*)d_in[4];   // [128,4096]
  const float* b2 = (const float*)d_in[5];   // [4096]
  float* out = (float*)d_out;                // [1,2,128,64]

  float* ws      = (float*)d_ws;
  float* QW1     = ws;                        // 32768 f
  float* KW1     = ws + 32768;                // 32768 f
  float* Hsum    = ws + 65536;                // 32768 f
  float* partial = ws + 98304;                // 8 * 16384 f  (total ~0.9 MB)

  k_qkw1<<<128, 256, 0, stream>>>(Q, K, W1, QW1, KW1);
  k_hsum<<<128, 256, 0, stream>>>(QW1, KW1, b1, Hsum);
  dim3 g3(16, 4, NC_CHUNKS);
  k_w2k<<<g3, 32, 0, stream>>>(K, W2, Hsum, partial);
  k_out<<<64, 256, 0, stream>>>(partial, K, b2, out);
}